// LocallyConnected_9964324127258
// MI455X (gfx1250) — compile-verified
//
#include <hip/hip_runtime.h>

// Locally-connected layer on gfx1250 (MI455X), fp32 via V_WMMA_F32_16X16X4_F32.
//
// out[n,o,l] = sum_k x_unf[n,l,k] * w[o,l,k] + bias[o,l]
//   N=64, O=64, L=62*62=3844, K=3*3*32=288
//
// Memory-bound: weight (283 MB) streams once from HBM; x (33.5 MB) and the
// 63 MB output live in the 192 MB L2. fp32 WMMA keeps full reference
// precision while comfortably exceeding the ~16 us HBM floor.

#define KH 3
#define KW 3
#define C_IN 32
#define H_IN 64
#define W_IN 64
#define N_B 64
#define O_OUT 64
#define HOUT 62
#define WOUT 62
#define L_LOC (HOUT * WOUT)     // 3844
#define K_DIM (KH * KW * C_IN)  // 288

#define KC 96                   // K chunk staged in LDS
#define NCHUNK (K_DIM / KC)     // 3
#define LDS_STRIDE 100          // KC + 4 pad -> conflict-free ds_load_b64

typedef float v2f __attribute__((ext_vector_type(2)));
typedef float v8f __attribute__((ext_vector_type(8)));

__global__ __launch_bounds__(512)
void lc_wmma_f32_kernel(const float* __restrict__ x,
                        const float* __restrict__ w,
                        const float* __restrict__ bias,
                        float* __restrict__ out)
{
    __shared__ float xs[N_B  * LDS_STRIDE];   // 25.0 KB  (x_unf slice, [n][k])
    __shared__ float ws[O_OUT * LDS_STRIDE];  // 25.0 KB  (weight slice, [o][k])

    const int l  = blockIdx.x;        // spatial location
    const int oy = l / WOUT;
    const int ox = l % WOUT;

    const int tid  = threadIdx.x;     // 512 threads = 16 wave32s
    const int lane = tid & 31;
    const int wv   = tid >> 5;        // wave id 0..15
    const int nt   = wv >> 2;         // batch-tile   (rows of 16 in N)
    const int ot   = wv & 3;          // o-tile       (cols of 16 in O)
    const int ml   = lane & 15;       // row/col index within 16
    const int khi  = lane >> 4;       // K half-select per WMMA layout

    // D layout: lane holds rows m = 8*khi + v, column n = ml.
    // bias depends only on the column (o) -> same value in all 8 acc slots,
    // so fold bias in by initializing the accumulator.
    const int ocol = ot * 16 + ml;    // output channel of this lane's column
    const float bv = bias[ocol * L_LOC + l];
    v8f acc;
#pragma unroll
    for (int i = 0; i < 8; ++i) acc[i] = bv;

    const float4* __restrict__ w4 = reinterpret_cast<const float4*>(w);

    for (int kc = 0; kc < NCHUNK; ++kc) {
        const int k0 = kc * KC;
        __syncthreads();   // previous chunk's compute done before overwrite

        // ---- weight chunk: 64 o x 96 k = 1536 float4, 3 per thread (coalesced)
#pragma unroll
        for (int it = 0; it < 3; ++it) {
            const int f  = tid + it * 512;        // 0..1535
            const int o  = f / (KC / 4);
            const int kq = f % (KC / 4);
            const int base = (o * L_LOC + l) * K_DIM + k0;   // multiple of 4
            const float4 v = w4[base / 4 + kq];
            *reinterpret_cast<float4*>(&ws[o * LDS_STRIDE + kq * 4]) = v;
            if (kc + 1 < NCHUNK) {
                // keep the 283 MB weight stream ahead: global_prefetch_b8
                __builtin_prefetch(w + base + KC + kq * 4, 0, 0);
            }
        }

        // ---- x_unf chunk: 64 n x 96 k scalars = 6144, 12 per thread
        //      k = c*9 + i*3 + j  (torch unfold order); gather served by L2
#pragma unroll
        for (int it = 0; it < 12; ++it) {
            const int f = tid + it * 512;
            const int n = f / KC;
            const int k = f % KC;
            const int kg = k + k0;
            const int c  = kg / (KH * KW);
            const int r  = kg % (KH * KW);
            const int i  = r / KW;
            const int j  = r % KW;
            xs[n * LDS_STRIDE + k] =
                x[((n * C_IN + c) * H_IN + (oy + i)) * W_IN + (ox + j)];
        }
        __syncthreads();

        // ---- 24 chained V_WMMA_F32_16X16X4_F32 steps over this chunk.
        // A(16x4) layout: lane 16*khi+m, vgpr v -> A[m][kk + 2*khi + v]
        // B(4x16) layout mirrors A -> one ds_load_b64 per operand per step.
        const float* xrow = &xs[(nt * 16 + ml) * LDS_STRIDE + 2 * khi];
        const float* wrow = &ws[(ot * 16 + ml) * LDS_STRIDE + 2 * khi];
#pragma unroll
        for (int kk = 0; kk < KC; kk += 4) {
            v2f a = *reinterpret_cast<const v2f*>(xrow + kk);
            v2f b = *reinterpret_cast<const v2f*>(wrow + kk);
            acc = __builtin_amdgcn_wmma_f32_16x16x4_f32(
                /*neg_a=*/false, a, /*neg_b=*/false, b,
                /*c_mod=*/(short)0, acc,
                /*reuse_a=*/false, /*reuse_b=*/false);
        }
    }

    // ---- store D: row m = nt*16 + 8*khi + v is the batch index, col = ocol.
    // (N,O,L) layout: stores are 4B-granular but land in L2 (63 MB output)
    // where adjacent-l blocks complete the cachelines.
    const int nbase = nt * 16 + 8 * khi;
#pragma unroll
    for (int v = 0; v < 8; ++v) {
        out[(nbase + v) * (O_OUT * L_LOC) + ocol * L_LOC + l] = acc[v];
    }
}

extern "C" void kernel_launch(void* const* d_in, const int* in_sizes, int n_in,
                              void* d_out, int out_size, void* d_ws, size_t ws_size,
                              hipStream_t stream) {
    const float* x    = (const float*)d_in[0];   // (64, 32, 64, 64)
    const float* w    = (const float*)d_in[1];   // (64, 3844, 288)
    const float* bias = (const float*)d_in[2];   // (64, 3844)
    float* out = (float*)d_out;                  // (64, 64, 62, 62)

    dim3 grid(L_LOC);   // one workgroup per spatial location
    dim3 block(512);    // 16 wave32s -> 4x4 grid of 16x16 WMMA tiles
    hipLaunchKernelGGL(lc_wmma_f32_kernel, grid, block, 0, stream,
                       x, w, bias, out);
}